// BiaffineCRF_91285234909742
// MI455X (gfx1250) — compile-verified
//
#include <hip/hip_runtime.h>
#include <hip/hip_bf16.h>

// Problem constants (from reference)
#define NCLUS 64      // K
#define NHEAD 8       // C
#define EMBED 512
#define BB    32
#define LL    256
#define HH    64      // EMBED/NHEAD
#define ROWS  (LL*NHEAD)   // 2048 rows of (row-major) X per batch

#define TSTRIDE 68    // padded row stride of T in LDS (row+8 -> +32 banks: halves disjoint)
#define ESTRIDE 72    // padded row stride of e1/e2 blocks in LDS (row+2 -> +16 banks)

typedef float v2f __attribute__((ext_vector_type(2)));
typedef float v8f __attribute__((ext_vector_type(8)));

// ---------------------------------------------------------------------------
// Phase 1: e1 = X @ W1^T + b1 ; e2 = X @ W2^T + b2  (per batch, X = (2048,64))
// f32 WMMA 16x16x4. One workgroup = 4 waves; wave w owns output col-tile w.
// ---------------------------------------------------------------------------
__global__ __launch_bounds__(128)
void biaffine_proj_kernel(const float* __restrict__ inputs,
                          const float* __restrict__ W1, const float* __restrict__ b1,
                          const float* __restrict__ W2, const float* __restrict__ b2,
                          float* __restrict__ e1, float* __restrict__ e2) {
    const int wg     = blockIdx.x;
    const int b      = wg >> 7;          // / (2048/16)
    const int tile_m = wg & 127;
    const int wave   = threadIdx.x >> 5;
    const int lane   = threadIdx.x & 31;
    const int half   = lane >> 4;        // 0: lanes 0-15, 1: lanes 16-31
    const int lid    = lane & 15;
    const int m0     = tile_m * 16;
    const int n0     = wave * 16;

    const float* X = inputs + (size_t)b * ROWS * HH;   // (2048,64) row-major
    v8f acc1 = {};
    v8f acc2 = {};

    #pragma unroll
    for (int k0 = 0; k0 < HH; k0 += 4) {
        const int ka = k0 + 2 * half;                  // K-pair for this half
        v2f a   = *(const v2f*)(X  + (size_t)(m0 + lid) * HH + ka);
        v2f bw1 = *(const v2f*)(W1 + (size_t)(n0 + lid) * HH + ka);
        v2f bw2 = *(const v2f*)(W2 + (size_t)(n0 + lid) * HH + ka);
        acc1 = __builtin_amdgcn_wmma_f32_16x16x4_f32(false, a, false, bw1,
                                                     (short)0, acc1, false, false);
        acc2 = __builtin_amdgcn_wmma_f32_16x16x4_f32(false, a, false, bw2,
                                                     (short)0, acc2, false, false);
    }

    const float bias1 = b1[n0 + lid];
    const float bias2 = b2[n0 + lid];
    float* E1 = e1 + (size_t)b * ROWS * NCLUS;
    float* E2 = e2 + (size_t)b * ROWS * NCLUS;
    #pragma unroll
    for (int v = 0; v < 8; ++v) {
        const int row = m0 + v + 8 * half;             // C/D layout, ISA 7.12.2
        E1[(size_t)row * NCLUS + n0 + lid] = acc1[v] + bias1;
        E2[(size_t)row * NCLUS + n0 + lid] = acc2[v] + bias2;
    }
}

// ---------------------------------------------------------------------------
// Phase 2: fused CRF forward scan. One workgroup (8 wave32) per sequence b.
// e1/e2 blocks double-buffered in LDS via global_load_async_to_lds_b128
// (ASYNCcnt), fetched one step ahead so the 4 KB copy overlaps the WMMA +
// logsumexp of the current step. T(64x64) built by 16 f32-WMMA tiles into
// padded LDS, then online logsumexp over k (4 partials per state j).
// ---------------------------------------------------------------------------
__global__ __launch_bounds__(256)
void biaffine_crf_scan_kernel(const float* __restrict__ e1, const float* __restrict__ e2,
                              const float* __restrict__ emissions,
                              const int* __restrict__ targets, const int* __restrict__ masks,
                              const float* __restrict__ start_tr,
                              const float* __restrict__ end_tr,
                              float* __restrict__ out) {
    const int b    = blockIdx.x;
    const int tid  = threadIdx.x;
    const int wave = tid >> 5;
    const int lane = tid & 31;
    const int half = lane >> 4;
    const int lid  = lane & 15;

    __shared__ float sT[NCLUS * TSTRIDE];         // padded transition matrix
    __shared__ float sE1[2][NHEAD * ESTRIDE];     // double-buffered e1 block
    __shared__ float sE2[2][NHEAD * ESTRIDE];     // double-buffered e2 block
    __shared__ float sS[NCLUS];                   // forward state
    __shared__ float sPm[4 * NCLUS];              // partial max
    __shared__ float sPs[4 * NCLUS];              // partial sum
    __shared__ float sNum;

    const float* emB = emissions + (size_t)b * LL * NCLUS;
    const int*   tgB = targets + (size_t)b * LL;
    const int*   mkB = masks   + (size_t)b * LL;
    const float* E1B = e1 + (size_t)b * ROWS * NCLUS;
    const float* E2B = e2 + (size_t)b * ROWS * NCLUS;

    // Per-thread async-copy slot: 16 bytes. Threads 0..127 -> e1, 128..255 -> e2.
    const int q   = tid & 127;
    const int qc  = q >> 4;            // c row (0..7)
    const int qx  = (q & 15) * 4;      // col within row (0,4,...,60)
    const int is2 = tid >> 7;

    auto issue_copy = [&](int l, int buf) {
        const float* gsrc = (is2 ? E2B : E1B)
                          + (size_t)l * NHEAD * NCLUS + qc * NCLUS + qx;
        float* ldst = (is2 ? &sE2[buf][0] : &sE1[buf][0]) + qc * ESTRIDE + qx;
        const unsigned loff = (unsigned)(unsigned long long)ldst;  // low 32b = LDS offset
        asm volatile("global_load_async_to_lds_b128 %0, %1, off"
                     :: "v"(loff), "v"(gsrc)
                     : "memory");
    };

    if (tid < NCLUS) sS[tid] = start_tr[tid] + emB[tid];
    float num = 0.0f;
    if (tid == 64) {
        const int t0 = tgB[0];
        num = start_tr[t0] + emB[t0];
    }
    issue_copy(1, 1);
    asm volatile("s_wait_asynccnt 0x0" ::: "memory");
    __syncthreads();

    for (int l = 1; l < LL; ++l) {
        const int cb = l & 1;
        if (l + 1 < LL) issue_copy(l + 1, cb ^ 1);   // prefetch next block (overlaps step)

        const float* E1s = &sE1[cb][0];
        const float* E2s = &sE2[cb][0];

        // --- T = e1^T(64x8) @ e2(8x64): two 16x16 tiles per wave, K=8 -> 2 WMMA ---
        #pragma unroll
        for (int t = 0; t < 2; ++t) {
            const int tt = wave * 2 + t;
            const int m0 = (tt >> 2) * 16;    // T row (k index)
            const int n0 = (tt & 3) * 16;     // T col (j index)
            v8f acc = {};
            #pragma unroll
            for (int c0 = 0; c0 < NHEAD; c0 += 4) {
                const int ca = c0 + 2 * half;
                v2f a, bb;
                a.x  = E1s[ca * ESTRIDE + m0 + lid];
                a.y  = E1s[(ca + 1) * ESTRIDE + m0 + lid];
                bb.x = E2s[ca * ESTRIDE + n0 + lid];
                bb.y = E2s[(ca + 1) * ESTRIDE + n0 + lid];
                acc = __builtin_amdgcn_wmma_f32_16x16x4_f32(false, a, false, bb,
                                                            (short)0, acc, false, false);
            }
            #pragma unroll
            for (int v = 0; v < 8; ++v)
                sT[(m0 + v + 8 * half) * TSTRIDE + n0 + lid] = acc[v];
        }
        __syncthreads();

        // --- online logsumexp: thread (part,j) reduces k in [16*part, 16*part+16) ---
        const int maskl = mkB[l];
        const int j    = tid & 63;
        const int part = tid >> 6;
        const int kb   = part * 16;
        float mval = -3.402823466e38f;
        float ssum = 0.0f;
        #pragma unroll 4
        for (int kk = 0; kk < 16; ++kk) {
            const float v = sS[kb + kk] + sT[(kb + kk) * TSTRIDE + j];
            if (v > mval) { ssum = ssum * __expf(mval - v) + 1.0f; mval = v; }
            else          { ssum += __expf(v - mval); }
        }
        sPm[part * NCLUS + j] = mval;
        sPs[part * NCLUS + j] = ssum;
        __syncthreads();

        if (tid < NCLUS) {
            const float m0v = sPm[tid],       m1 = sPm[64 + tid];
            const float m2  = sPm[128 + tid], m3 = sPm[192 + tid];
            const float M = fmaxf(fmaxf(m0v, m1), fmaxf(m2, m3));
            const float S = sPs[tid]       * __expf(m0v - M)
                          + sPs[64 + tid]  * __expf(m1 - M)
                          + sPs[128 + tid] * __expf(m2 - M)
                          + sPs[192 + tid] * __expf(m3 - M);
            const float nxt = emB[(size_t)l * NCLUS + tid] + M + __logf(S);
            if (maskl) sS[tid] = nxt;
        }
        if (tid == 64 && maskl) {           // gold-path numerator (sT still valid)
            const int tp = tgB[l - 1], tc = tgB[l];
            num += sT[tp * TSTRIDE + tc] + emB[(size_t)l * NCLUS + tc];
        }
        asm volatile("s_wait_asynccnt 0x0" ::: "memory");  // next block landed
        __syncthreads();
    }

    // --- end transitions + final reductions ---
    if (tid == 64) {
        int send = 0;
        for (int l = 0; l < LL; ++l) send += (mkB[l] != 0);
        const int lastTag = tgB[send - 1];
        sNum = num + end_tr[lastTag];
    }
    if (tid < NCLUS) sPm[tid] = sS[tid] + end_tr[tid];
    __syncthreads();

    if (tid == 0) {
        float M = sPm[0];
        for (int k = 1; k < NCLUS; ++k) M = fmaxf(M, sPm[k]);
        float S = 0.0f;
        for (int k = 0; k < NCLUS; ++k) S += __expf(sPm[k] - M);
        out[b] = sNum - (M + __logf(S));
    }
}

// ---------------------------------------------------------------------------
extern "C" void kernel_launch(void* const* d_in, const int* in_sizes, int n_in,
                              void* d_out, int out_size, void* d_ws, size_t ws_size,
                              hipStream_t stream) {
    const float* inputs    = (const float*)d_in[0];
    const float* emissions = (const float*)d_in[1];
    const int*   targets   = (const int*)d_in[2];
    const int*   masks     = (const int*)d_in[3];
    const float* W1        = (const float*)d_in[4];
    const float* b1        = (const float*)d_in[5];
    const float* W2        = (const float*)d_in[6];
    const float* b2        = (const float*)d_in[7];
    const float* start_tr  = (const float*)d_in[8];
    const float* end_tr    = (const float*)d_in[9];
    float* out = (float*)d_out;

    // Workspace: e1, e2 each B*2048*64 f32 (16.8 MB) -> 33.5 MB total.
    float* e1 = (float*)d_ws;
    float* e2 = e1 + (size_t)BB * ROWS * NCLUS;

    biaffine_proj_kernel<<<dim3(BB * (ROWS / 16)), dim3(128), 0, stream>>>(
        inputs, W1, b1, W2, b2, e1, e2);
    biaffine_crf_scan_kernel<<<dim3(BB), dim3(256), 0, stream>>>(
        e1, e2, emissions, targets, masks, start_tr, end_tr, out);
}